// AggregationDecoder_63367947485216
// MI455X (gfx1250) — compile-verified
//
#include <hip/hip_runtime.h>
#include <hip/hip_bf16.h>

// Problem constants (match reference)
#define NUM_MESH_NODES 40962
#define NUM_GRID_NODES 65536
#define EMBED          256
#define NUM_EDGES      262144
#define BATCH          2
#define MAXD           64   // per-bucket LDS staging capacity (P(deg>64) ~ 1e-60 for Poisson(4))

typedef float    v4f __attribute__((ext_vector_type(4)));
typedef int      v2i __attribute__((ext_vector_type(2)));
typedef unsigned v4u __attribute__((ext_vector_type(4)));

// d_ws layout (as unsigned words):
//   deg     : [0, 65536)
//   offs    : [65536, 131073)   (65537 entries: exclusive scan + total)
//   cursor  : [131080, 196616)  (running fill cursors)
//   senders : [196616, 458760)  (binned sender ids, NUM_EDGES entries)
// total ~1.76 MB

// ---------------------------------------------------------------------------
// Kernel 1: zero degree counters (B128 stores)
// ---------------------------------------------------------------------------
__global__ __launch_bounds__(256) void zero_deg_kernel(v4u* __restrict__ deg4, int n4) {
    int t = blockIdx.x * 256 + threadIdx.x;
    if (t < n4) {
        v4u z = {0u, 0u, 0u, 0u};
        deg4[t] = z;
    }
}

// ---------------------------------------------------------------------------
// Kernel 2: degree histogram (non-returning u32 atomics, paired B64 edge load)
// ---------------------------------------------------------------------------
__global__ __launch_bounds__(256) void deg_kernel(const v2i* __restrict__ eidx2,
                                                  unsigned* __restrict__ deg) {
    int e = blockIdx.x * 256 + threadIdx.x;
    if (e < NUM_EDGES) {
        v2i p = eidx2[e];           // {send, recv} in one B64 load
        atomicAdd(&deg[p.y], 1u);
    }
}

// ---------------------------------------------------------------------------
// Kernel 3: exclusive prefix scan of 65536 degrees, single 1024-thread block.
// Each thread owns 64 consecutive counters. Writes offs[] and cursor[].
// ---------------------------------------------------------------------------
__global__ __launch_bounds__(1024) void scan_kernel(const unsigned* __restrict__ deg,
                                                    unsigned* __restrict__ offs,
                                                    unsigned* __restrict__ cursor) {
    __shared__ unsigned part[1024];
    const int t = threadIdx.x;
    const int base = t * 64;

    unsigned sum = 0;
    for (int i = 0; i < 64; ++i) sum += deg[base + i];
    part[t] = sum;
    __syncthreads();

    // Hillis-Steele inclusive scan over 1024 partials
    for (int off = 1; off < 1024; off <<= 1) {
        unsigned v   = part[t];
        unsigned add = (t >= off) ? part[t - off] : 0u;
        __syncthreads();
        part[t] = v + add;
        __syncthreads();
    }

    unsigned run = (t == 0) ? 0u : part[t - 1];   // exclusive base for this thread
    for (int i = 0; i < 64; ++i) {
        unsigned d = deg[base + i];
        offs[base + i]   = run;
        cursor[base + i] = run;
        run += d;
    }
    if (t == 1023) offs[NUM_GRID_NODES] = run;    // == NUM_EDGES
}

// ---------------------------------------------------------------------------
// Kernel 4: bin sender ids by receiver (returning u32 atomics on cursors)
// ---------------------------------------------------------------------------
__global__ __launch_bounds__(256) void fill_kernel(const v2i* __restrict__ eidx2,
                                                   unsigned* __restrict__ cursor,
                                                   unsigned* __restrict__ senders) {
    int e = blockIdx.x * 256 + threadIdx.x;
    if (e < NUM_EDGES) {
        v2i p = eidx2[e];                              // {send, recv}
        unsigned pos = atomicAdd(&cursor[p.y], 1u);    // returning u32 atomic
        senders[pos] = (unsigned)p.x;
    }
}

// ---------------------------------------------------------------------------
// Kernel 5: one wave32 per grid node, both batches. Stage + sort the bucket
// in LDS (deterministic accumulation order across replays), accumulate
// gathered feature rows (RT loads -> keep 84MB table L2-resident), and store
// the write-once 134MB output with NT hints so it does not evict the table.
// ---------------------------------------------------------------------------
__global__ __launch_bounds__(256) void gather_kernel(const float* __restrict__ feat,
                                                     const unsigned* __restrict__ offs,
                                                     const unsigned* __restrict__ senders,
                                                     float* __restrict__ out) {
    __shared__ int buf[8][MAXD];                 // 8 waves/block * 64 entries
    const int tid  = blockIdx.x * 256 + threadIdx.x;
    const int g    = tid >> 5;                   // grid node = global wave id
    const int lw   = threadIdx.x >> 5;           // wave within block
    const int lane = threadIdx.x & 31;

    const unsigned beg = offs[g];
    const unsigned end = offs[g + 1];
    const int n = (int)(end - beg);

    // cooperative bucket load into LDS
    for (int k = lane; k < n && k < MAXD; k += 32)
        buf[lw][k] = (int)senders[beg + k];
    __syncthreads();

    // lane 0: insertion sort (n is tiny) -> deterministic float add order
    if (lane == 0 && n <= MAXD) {
        for (int i = 1; i < n; ++i) {
            int key = buf[lw][i];
            int j = i - 1;
            while (j >= 0 && buf[lw][j] > key) { buf[lw][j + 1] = buf[lw][j]; --j; }
            buf[lw][j + 1] = key;
        }
    }
    __syncthreads();

    // accumulate: each lane covers channels [lane*4, lane*4+4) and +128, per batch
    v4f a00 = {0.f, 0.f, 0.f, 0.f};  // batch 0, low half
    v4f a01 = {0.f, 0.f, 0.f, 0.f};  // batch 0, high half
    v4f a10 = {0.f, 0.f, 0.f, 0.f};  // batch 1, low half
    v4f a11 = {0.f, 0.f, 0.f, 0.f};  // batch 1, high half

    const float* base0 = feat;
    const float* base1 = feat + (size_t)NUM_MESH_NODES * EMBED;

    for (int k = 0; k < n; ++k) {
        int s = (k < MAXD) ? buf[lw][k] : (int)senders[beg + k];

        // lookahead prefetch of next sender row (global_prefetch_b8)
        if (k + 1 < n && k + 1 < MAXD) {
            int s2 = buf[lw][k + 1];
            __builtin_prefetch(base0 + (size_t)s2 * EMBED + lane * 4, 0, 3);
            __builtin_prefetch(base1 + (size_t)s2 * EMBED + lane * 4, 0, 3);
        }

        const float* r0 = base0 + (size_t)s * EMBED;
        const float* r1 = base1 + (size_t)s * EMBED;
        a00 += *(const v4f*)(r0 + lane * 4);        // global_load_b128, RT (reused rows)
        a01 += *(const v4f*)(r0 + 128 + lane * 4);
        a10 += *(const v4f*)(r1 + lane * 4);
        a11 += *(const v4f*)(r1 + 128 + lane * 4);
    }

    const float invd = 1.0f / fmaxf((float)n, 1.0f);
    a00 *= invd; a01 *= invd; a10 *= invd; a11 *= invd;

    float* d0 = out + (size_t)g * EMBED;                                   // batch 0 row
    float* d1 = out + (size_t)NUM_GRID_NODES * EMBED + (size_t)g * EMBED;  // batch 1 row
    // write-once stream: non-temporal B128 stores (do not evict feature table)
    __builtin_nontemporal_store(a00, (v4f*)(d0 + lane * 4));
    __builtin_nontemporal_store(a01, (v4f*)(d0 + 128 + lane * 4));
    __builtin_nontemporal_store(a10, (v4f*)(d1 + lane * 4));
    __builtin_nontemporal_store(a11, (v4f*)(d1 + 128 + lane * 4));
}

// ---------------------------------------------------------------------------
// Host launcher (graph-capture safe: only kernel launches on `stream`)
// ---------------------------------------------------------------------------
extern "C" void kernel_launch(void* const* d_in, const int* in_sizes, int n_in,
                              void* d_out, int out_size, void* d_ws, size_t ws_size,
                              hipStream_t stream) {
    const float* feat  = (const float*)d_in[0];   // (B, M, C) f32
    const v2i*   eidx2 = (const v2i*)d_in[1];     // (E, 2) int, paired
    float*       out   = (float*)d_out;           // (B, G, C) f32

    unsigned* w       = (unsigned*)d_ws;
    unsigned* deg     = w;                        // 65536
    unsigned* offs    = w + 65536;                // 65537
    unsigned* cursor  = w + 131080;               // 65536 (padded start)
    unsigned* senders = w + 196616;               // 262144

    // 1) zero degree counters
    zero_deg_kernel<<<64, 256, 0, stream>>>((v4u*)deg, NUM_GRID_NODES / 4);

    // 2) degree histogram
    deg_kernel<<<NUM_EDGES / 256, 256, 0, stream>>>(eidx2, deg);

    // 3) exclusive scan -> offsets + cursors (single block)
    scan_kernel<<<1, 1024, 0, stream>>>(deg, offs, cursor);

    // 4) bin sender ids by receiver
    fill_kernel<<<NUM_EDGES / 256, 256, 0, stream>>>(eidx2, cursor, senders);

    // 5) gather + normalize + NT store (one wave per grid node, both batches)
    gather_kernel<<<(NUM_GRID_NODES * 32) / 256, 256, 0, stream>>>(feat, offs, senders, out);
}